// HeterogeneousExposure_78073915507114
// MI455X (gfx1250) — compile-verified
//
#include <hip/hip_runtime.h>
#include <hip/hip_bf16.h>
#include <stdint.h>

typedef __attribute__((ext_vector_type(2))) float v2f;
typedef __attribute__((ext_vector_type(8))) float v8f;

#define HID    64
#define KDIM   132          // 2*HID + sim, padded to multiple of 4 for K=4 WMMA steps
#define TILE   16           // edges per wave (one WMMA M dimension)
#define WAVES  4            // waves per block

// ---------------------------------------------------------------------------
// Per-edge kernel: async-DMA node rows -> LDS (GLOBAL_LOAD_ASYNC_TO_LDS_B128),
// compute Gaussian sim, run the [16x132]x[132x16] MLP layer on the matrix
// pipe (33x v_wmma_f32_16x16x4_f32), ELU + 16->1 layer via half-wave shuffle
// reduction, sigmoid, then atomic scatter of w*h_src into out[dst] and w into
// wsum[dst].
// ---------------------------------------------------------------------------
__global__ __launch_bounds__(WAVES * 32) void edge_mlp_scatter(
    const float* __restrict__ h,        // [n, 64]
    const long long* __restrict__ ei,   // [2, E]  row0 = src, row1 = dst
    const float* __restrict__ W1,       // [129, 16]
    const float* __restrict__ b1,       // [16]
    const float* __restrict__ W2,       // [16]
    const float* __restrict__ b2,       // [1]
    float* __restrict__ out,            // [n, 64] accumulators (pre-zeroed)
    float* __restrict__ wsum,           // [n]     accumulators (pre-zeroed)
    int E, int ntiles)
{
    __shared__ float     s_w1t[16][KDIM];            // W1 transposed+padded: [N][K]
    __shared__ float     s_feat[WAVES][TILE][KDIM];  // per-edge feature rows
    __shared__ float     s_ew[WAVES][TILE];          // per-edge weights
    __shared__ long long s_dst[WAVES][TILE];
    __shared__ float     s_b1[16];
    __shared__ float     s_w2[16];

    const int tid  = threadIdx.x;
    const int lane = tid & 31;
    const int wv   = tid >> 5;

    // Stage W1^T (padded) + bias vectors into LDS once per block.
    for (int i = tid; i < 16 * KDIM; i += blockDim.x) {
        int nn = i / KDIM, k = i % KDIM;
        s_w1t[nn][k] = (k < 2 * HID + 1) ? W1[k * 16 + nn] : 0.0f;
    }
    if (tid < 16) { s_b1[tid] = b1[tid]; s_w2[tid] = W2[tid]; }
    const float b2v = b2[0];
    __syncthreads();

    const int  tile       = blockIdx.x * WAVES + wv;
    const bool tile_valid = tile < ntiles;
    const int  e_base     = tile * TILE;

    // ---- Gather: 2 lanes per edge; each DMAs one 64-float node row straight
    //      from global memory into LDS (no VGPR round-trip), 16x async b128.
    {
        const int  m   = lane >> 1;
        const int  hh  = lane & 1;                 // 0 -> h_dst (feat[0:64]), 1 -> h_src (feat[64:128])
        const int  eid = e_base + m;
        const int  ec  = (tile_valid && eid < E) ? eid : 0;
        const long long row = hh ? ei[ec] : ei[(size_t)E + ec];
        const float* gsrc = h + (size_t)row * HID;
        // Generic pointers to LDS carry the LDS byte offset in their low 32
        // bits -> usable as the VDST LDS address.
        unsigned ldsoff = (unsigned)(uintptr_t)(const void*)&s_feat[wv][m][hh * HID];
        #pragma unroll
        for (int k = 0; k < HID / 4; ++k) {
            asm volatile("global_load_async_to_lds_b128 %0, %1, off offset:%2"
                         :
                         : "v"(ldsoff), "v"(gsrc), "i"(k * 16)
                         : "memory");
        }
        if (hh == 0) s_dst[wv][m] = row;
    }
    // All async DMAs must land in LDS before any wave reads the tile.
    asm volatile("s_wait_asynccnt 0x0" ::: "memory");
    __syncthreads();

    // ---- Gaussian similarity: 2 lanes per edge, half each, combine via shuffle
    {
        const int m = lane >> 1, hh = lane & 1;
        const float* fd = &s_feat[wv][m][hh * 32];
        const float* fs = &s_feat[wv][m][HID + hh * 32];
        float s = 0.0f;
        #pragma unroll
        for (int k = 0; k < 32; ++k) { float d = fd[k] - fs[k]; s += d * d; }
        s += __shfl_xor(s, 1, 32);
        if (hh == 0) {
            s_feat[wv][m][128] = __expf(-s / (2.0f * HID));
            s_feat[wv][m][129] = 0.0f;
            s_feat[wv][m][130] = 0.0f;
            s_feat[wv][m][131] = 0.0f;
        }
    }
    __syncthreads();

    // ---- Layer 1 on the matrix pipe: 33 x v_wmma_f32_16x16x4_f32
    // A/B fragment layout: lane l holds (row l%16, K = 4c + 2*(l/16) + {0,1})
    v8f acc = {};
    {
        const int rc   = lane & 15;          // A row (edge) == B col (hidden unit)
        const int koff = (lane >> 4) * 2;
        #pragma unroll
        for (int c = 0; c < KDIM / 4; ++c) {
            v2f a = *(const v2f*)&s_feat[wv][rc][c * 4 + koff];
            v2f b = *(const v2f*)&s_w1t[rc][c * 4 + koff];
            acc = __builtin_amdgcn_wmma_f32_16x16x4_f32(
                false, a, false, b, (short)0, acc, false, false);
        }
    }

    // ---- bias + ELU + layer 2 (16->1) + sigmoid.
    // C layout: lane l, vgpr j -> (M = j + 8*(l/16), N = l%16)
    {
        const int   nn   = lane & 15;
        const int   half = lane >> 4;
        const float w2v  = s_w2[nn];
        const float b1v  = s_b1[nn];
        float myw[8];
        #pragma unroll
        for (int j = 0; j < 8; ++j) {
            float x    = acc[j] + b1v;
            float hdnv = x > 0.0f ? x : (__expf(x) - 1.0f);   // ELU
            float t    = hdnv * w2v;
            t += __shfl_xor(t, 1, 32);
            t += __shfl_xor(t, 2, 32);
            t += __shfl_xor(t, 4, 32);
            t += __shfl_xor(t, 8, 32);                         // sum over N in half-wave
            myw[j] = 1.0f / (1.0f + __expf(-(t + b2v)));       // sigmoid
        }
        if (nn == 0) {
            #pragma unroll
            for (int j = 0; j < 8; ++j) s_ew[wv][half * 8 + j] = myw[j];
        }
    }
    __syncthreads();

    // ---- Scatter: 2 lanes per edge, 32 atomic f32 adds each + wsum add
    {
        const int m   = lane >> 1;
        const int hh  = lane & 1;
        const int eid = e_base + m;
        if (tile_valid && eid < E) {
            const float     wE   = s_ew[wv][m];
            const long long drow = s_dst[wv][m];
            float*       op = out + (size_t)drow * HID + hh * 32;
            const float* sp = &s_feat[wv][m][HID + hh * 32];   // h_src half
            #pragma unroll
            for (int k = 0; k < 32; ++k) unsafeAtomicAdd(&op[k], sp[k] * wE);
            if (hh == 0) unsafeAtomicAdd(&wsum[drow], wE);
        }
    }
}

// ---------------------------------------------------------------------------
// out[i][:] /= max(wsum[i], 1e-8)
// ---------------------------------------------------------------------------
__global__ __launch_bounds__(256) void finalize_div(
    const float* __restrict__ wsum, float* __restrict__ out, long long total)
{
    long long i = (long long)blockIdx.x * blockDim.x + threadIdx.x;
    if (i < total) {
        float ws = wsum[i >> 6];              // HID = 64
        ws = ws > 1e-8f ? ws : 1e-8f;
        out[i] = out[i] / ws;
    }
}

extern "C" void kernel_launch(void* const* d_in, const int* in_sizes, int n_in,
                              void* d_out, int out_size, void* d_ws, size_t ws_size,
                              hipStream_t stream) {
    const float*     h  = (const float*)d_in[0];
    const long long* ei = (const long long*)d_in[1];     // int64 [2, E]
    const float*     W1 = (const float*)d_in[2];
    const float*     b1 = (const float*)d_in[3];
    const float*     W2 = (const float*)d_in[4];
    const float*     b2 = (const float*)d_in[5];
    float* out  = (float*)d_out;
    float* wsum = (float*)d_ws;

    const int n = in_sizes[0] / HID;
    const int E = in_sizes[1] / 2;

    hipMemsetAsync(out,  0, (size_t)n * HID * sizeof(float), stream);
    hipMemsetAsync(wsum, 0, (size_t)n * sizeof(float), stream);

    const int ntiles  = (E + TILE - 1) / TILE;
    const int nblocks = (ntiles + WAVES - 1) / WAVES;
    edge_mlp_scatter<<<nblocks, WAVES * 32, 0, stream>>>(
        h, ei, W1, b1, W2, b2, out, wsum, E, ntiles);

    const long long total = (long long)n * HID;
    finalize_div<<<(int)((total + 255) / 256), 256, 0, stream>>>(wsum, out, total);
}